// conditioning_layer_773094113350
// MI455X (gfx1250) — compile-verified
//
#include <hip/hip_runtime.h>

#define BATCH 32
#define CDIM  256
#define HWDIM 9216          // 96*96
#define KRANK 2764          // int(0.3 * 9216)

typedef __attribute__((ext_vector_type(2))) float v2f;
typedef __attribute__((ext_vector_type(8))) float v8f;

// ---------------------------------------------------------------------------
// Pass 1: x[b,s] = sum_c z[b,c,s] * phi_w[c] + phi_b   (streams z once)
// grid (HW/1024, B), block 256; each thread owns 4 consecutive s (float4)
// ---------------------------------------------------------------------------
__global__ __launch_bounds__(256) void k_project(const float* __restrict__ z,
                                                 const float* __restrict__ phi_w,
                                                 const float* __restrict__ phi_b,
                                                 float* __restrict__ x) {
    const int b  = blockIdx.y;
    const int s0 = blockIdx.x * 1024 + threadIdx.x * 4;
    const float* zb = z + (size_t)b * CDIM * HWDIM + s0;
    float4 acc = make_float4(0.f, 0.f, 0.f, 0.f);
#pragma unroll 8
    for (int c = 0; c < CDIM; ++c) {
        const float  w = phi_w[c];                       // wave-uniform -> s_load
        const float4 v = *(const float4*)(zb + (size_t)c * HWDIM);
        acc.x += w * v.x; acc.y += w * v.y; acc.z += w * v.z; acc.w += w * v.w;
    }
    const float pb = phi_b[0];
    acc.x += pb; acc.y += pb; acc.z += pb; acc.w += pb;
    *(float4*)(x + (size_t)b * HWDIM + s0) = acc;
}

// ---------------------------------------------------------------------------
// Pass 2: per-batch exact k-th largest of x[b,:] via MSB-first radix select
// one block per batch, keys in LDS, 8-bit digits, LDS histogram
// ---------------------------------------------------------------------------
__device__ __forceinline__ unsigned f2ord(float f) {
    unsigned u = __float_as_uint(f);
    return (u & 0x80000000u) ? ~u : (u | 0x80000000u);
}
__device__ __forceinline__ float ord2f(unsigned k) {
    unsigned u = (k & 0x80000000u) ? (k & 0x7FFFFFFFu) : ~k;
    return __uint_as_float(u);
}

__global__ __launch_bounds__(256) void k_select(const float* __restrict__ x,
                                                float* __restrict__ thresh) {
    __shared__ unsigned skeys[HWDIM];   // 36 KB
    __shared__ unsigned hist[256];
    __shared__ unsigned s_prefix, s_rem;
    const int b   = blockIdx.x;
    const int tid = threadIdx.x;

    for (int i = tid; i < HWDIM; i += 256)
        skeys[i] = f2ord(x[(size_t)b * HWDIM + i]);
    if (tid == 0) { s_prefix = 0u; s_rem = KRANK; }
    __syncthreads();

    for (int shift = 24; shift >= 0; shift -= 8) {
        hist[tid] = 0u;
        __syncthreads();
        const unsigned prefix = s_prefix;
        const unsigned hiMask = (shift == 24) ? 0u : (0xFFFFFFFFu << (shift + 8));
        for (int i = tid; i < HWDIM; i += 256) {
            const unsigned key = skeys[i];
            if ((key & hiMask) == prefix)
                atomicAdd(&hist[(key >> shift) & 255u], 1u);
        }
        __syncthreads();
        if (tid == 0) {
            unsigned rem = s_rem;
            int dsel = 0;
            for (int d = 255; d >= 0; --d) {
                const unsigned c = hist[d];
                if (rem > c) { rem -= c; } else { dsel = d; break; }
            }
            s_prefix = prefix | ((unsigned)dsel << shift);
            s_rem    = rem;
        }
        __syncthreads();
    }
    if (tid == 0) thresh[b] = ord2f(s_prefix);
}

// ---------------------------------------------------------------------------
// Pass 3: gap[b,c] = mean_s z[b,c,s] * (x[b,s] > thresh[b])   (streams z again)
// grid (C/32, B), block 256 = 8 waves; mask built once in LDS per block,
// each wave streams 4 channel rows (B128 loads) and wave32-reduces.
// ---------------------------------------------------------------------------
__global__ __launch_bounds__(256) void k_gap(const float* __restrict__ z,
                                             const float* __restrict__ x,
                                             const float* __restrict__ thresh,
                                             float* __restrict__ gap) {
    __shared__ float smask[HWDIM];      // 36 KB
    const int   b   = blockIdx.y;
    const int   tid = threadIdx.x;
    const float th  = thresh[b];

    for (int i = tid * 4; i < HWDIM; i += 1024) {
        const float4 xv = *(const float4*)(x + (size_t)b * HWDIM + i);
        float4 m;
        m.x = xv.x > th ? 1.f : 0.f;
        m.y = xv.y > th ? 1.f : 0.f;
        m.z = xv.z > th ? 1.f : 0.f;
        m.w = xv.w > th ? 1.f : 0.f;
        *(float4*)&smask[i] = m;
    }
    __syncthreads();

    const int wave  = tid >> 5;
    const int lane  = tid & 31;
    const int cbase = blockIdx.x * 32 + wave * 4;
    const float4* mp = (const float4*)smask;

    for (int cc = 0; cc < 4; ++cc) {
        const int c = cbase + cc;
        const float4* zp = (const float4*)(z + (size_t)b * CDIM * HWDIM + (size_t)c * HWDIM);
        float acc = 0.f;
#pragma unroll 4
        for (int i = lane; i < HWDIM / 4; i += 32) {
            const float4 v = zp[i];
            const float4 m = mp[i];
            acc += v.x * m.x + v.y * m.y + v.z * m.z + v.w * m.w;
        }
#pragma unroll
        for (int off = 16; off > 0; off >>= 1)
            acc += __shfl_down(acc, off, 32);
        if (lane == 0) gap[b * CDIM + c] = acc * (1.f / HWDIM);
    }
}

// ---------------------------------------------------------------------------
// Pass 4: out[b,j] = sum_c gap[b,c] * mlp_w[j,c] + mlp_b[j]
// 32x256 = (32x256)x(256x256)^T via V_WMMA_F32_16X16X4_F32.
// One wave per 16x16 output tile; 64 K-steps of 4.
// A 16x4 layout: lanes 0-15 hold K=0,1 (VGPR0,1); lanes 16-31 hold K=2,3.
// B 4x16 layout mirrors it: lanes 0-15 rows K=0,1; lanes 16-31 rows K=2,3.
// ---------------------------------------------------------------------------
__global__ __launch_bounds__(256) void k_mlp_wmma(const float* __restrict__ gap,
                                                  const float* __restrict__ mlp_w,
                                                  const float* __restrict__ mlp_b,
                                                  float* __restrict__ out) {
    const int wave = threadIdx.x >> 5;
    const int lane = threadIdx.x & 31;
    const int tile = blockIdx.x * 8 + wave;   // 0..31
    const int mi   = tile >> 4;               // 0..1   (row tile)
    const int nj   = tile & 15;               // 0..15  (col tile)
    const int l16  = lane & 15;
    const int hi   = (lane >> 4) & 1;
    const int arow = mi * 16 + l16;           // gap row this lane feeds
    const int ncol = nj * 16 + l16;           // out col this lane owns
    const int koff = hi * 2;

    v8f acc = {};
#pragma unroll 8
    for (int k0 = 0; k0 < CDIM; k0 += 4) {
        v2f a, bm;
        a[0]  = gap[arow * CDIM + k0 + koff];
        a[1]  = gap[arow * CDIM + k0 + koff + 1];
        bm[0] = mlp_w[ncol * CDIM + k0 + koff];       // B[k][n] = mlp_w[n][k]
        bm[1] = mlp_w[ncol * CDIM + k0 + koff + 1];
        acc = __builtin_amdgcn_wmma_f32_16x16x4_f32(
            /*neg_a=*/false, a, /*neg_b=*/false, bm,
            /*c_mod=*/(short)0, acc, /*reuse_a=*/false, /*reuse_b=*/false);
    }

    const float bias = mlp_b[ncol];
#pragma unroll
    for (int r = 0; r < 8; ++r) {
        const int row = mi * 16 + hi * 8 + r;         // C/D VGPR r <-> M=r / M=8+r
        out[row * CDIM + ncol] = acc[r] + bias;
    }
}

// ---------------------------------------------------------------------------
extern "C" void kernel_launch(void* const* d_in, const int* in_sizes, int n_in,
                              void* d_out, int out_size, void* d_ws, size_t ws_size,
                              hipStream_t stream) {
    const float* z     = (const float*)d_in[0];
    const float* phi_w = (const float*)d_in[1];
    const float* phi_b = (const float*)d_in[2];
    const float* mlp_w = (const float*)d_in[3];
    const float* mlp_b = (const float*)d_in[4];
    float* out = (float*)d_out;

    float* x      = (float*)d_ws;                  // B*HW floats (1.18 MB)
    float* thresh = x + (size_t)BATCH * HWDIM;     // B floats
    float* gap    = thresh + BATCH;                // B*C floats

    k_project<<<dim3(HWDIM / 1024, BATCH), 256, 0, stream>>>(z, phi_w, phi_b, x);
    k_select <<<BATCH, 256, 0, stream>>>(x, thresh);
    k_gap    <<<dim3(CDIM / 32, BATCH), 256, 0, stream>>>(z, x, thresh, gap);
    k_mlp_wmma<<<4, 256, 0, stream>>>(gap, mlp_w, mlp_b, out);
}